// CustomMultiheadAttention_10058813407800
// MI455X (gfx1250) — compile-verified
//
#include <hip/hip_runtime.h>
#include <hip/hip_bf16.h>

typedef float v2f __attribute__((ext_vector_type(2)));
typedef float v8f __attribute__((ext_vector_type(8)));

#define LQ 2048
#define NB 2
#define EM 1024
#define HD 64
#define NH 16
#define NBH 32              // NB * NH
#define LDSS 68             // padded LDS row stride (floats), multiple of 4

__device__ __forceinline__ v8f wmma4(v2f a, v2f b, v8f c) {
  // D = A(16x4 f32) * B(4x16 f32) + C(16x16 f32)
  return __builtin_amdgcn_wmma_f32_16x16x4_f32(
      /*neg_a=*/false, a, /*neg_b=*/false, b,
      /*c_mod=*/(short)0, c, /*reuse_a=*/false, /*reuse_b=*/false);
}

__device__ __forceinline__ v8f v8zero() {
  v8f z = {0.f,0.f,0.f,0.f,0.f,0.f,0.f,0.f};
  return z;
}

// ---------------------------------------------------------------------------
// Kernel 1: projection  dst_hm[b=(n*16+h)][l][d] = (X @ W^T)[m=l*2+n][e=h*64+d] + bias[e]
// X: [4096,1024] row-major, W: [1024,1024] row-major (rows = output features)
// ---------------------------------------------------------------------------
__global__ __launch_bounds__(128)
void proj_kernel(const float* __restrict__ X, const float* __restrict__ W,
                 const float* __restrict__ bias, float* __restrict__ dst_hm) {
  __shared__ __align__(16) float sA[64 * LDSS];
  __shared__ __align__(16) float sB[64 * LDSS];

  const int tid  = threadIdx.x;
  const int lane = tid & 31;
  const int wv   = tid >> 5;        // wave 0..3
  const int l15  = lane & 15;
  const int hi   = lane >> 4;       // 0/1
  const int m0   = blockIdx.x * 64;
  const int n0   = blockIdx.y * 64;
  const int wm   = (wv >> 1) * 32;  // wave M offset in tile
  const int wn   = (wv & 1) * 32;   // wave N offset in tile

  v8f acc[2][2];
  acc[0][0] = v8zero(); acc[0][1] = v8zero();
  acc[1][0] = v8zero(); acc[1][1] = v8zero();

  const int row   = tid >> 1;        // 0..63
  const int cbase = (tid & 1) * 16;  // 0/16

  for (int k0 = 0; k0 < EM; k0 += 32) {
    __syncthreads();
#pragma unroll
    for (int i = 0; i < 4; ++i) {
      const int c = cbase + i * 4;
      *(float4*)&sA[row * LDSS + c] = *(const float4*)&X[(m0 + row) * EM + k0 + c];
      *(float4*)&sB[row * LDSS + c] = *(const float4*)&W[(n0 + row) * EM + k0 + c];
    }
    if (k0 + 32 < EM) {  // hint next chunk into cache (global_prefetch_b8)
      __builtin_prefetch(&X[(m0 + row) * EM + k0 + 32 + cbase], 0, 3);
      __builtin_prefetch(&W[(n0 + row) * EM + k0 + 32 + cbase], 0, 3);
    }
    __syncthreads();
#pragma unroll
    for (int kk = 0; kk < 32; kk += 4) {
      v2f a0 = *(const v2f*)&sA[(wm      + l15) * LDSS + kk + 2 * hi];
      v2f a1 = *(const v2f*)&sA[(wm + 16 + l15) * LDSS + kk + 2 * hi];
      v2f b0 = *(const v2f*)&sB[(wn      + l15) * LDSS + kk + 2 * hi];
      v2f b1 = *(const v2f*)&sB[(wn + 16 + l15) * LDSS + kk + 2 * hi];
      acc[0][0] = wmma4(a0, b0, acc[0][0]);
      acc[0][1] = wmma4(a0, b1, acc[0][1]);
      acc[1][0] = wmma4(a1, b0, acc[1][0]);
      acc[1][1] = wmma4(a1, b1, acc[1][1]);
    }
  }

#pragma unroll
  for (int im = 0; im < 2; ++im) {
#pragma unroll
    for (int in = 0; in < 2; ++in) {
      const int e  = n0 + wn + in * 16 + l15;
      const float bv = bias[e];
      const int h = e >> 6, d = e & 63;
#pragma unroll
      for (int r = 0; r < 8; ++r) {
        const int m = m0 + wm + im * 16 + hi * 8 + r;
        const int l = m >> 1, nn = m & 1;
        dst_hm[((nn * NH + h) * LQ + l) * HD + d] = acc[im][in][r] + bv;
      }
    }
  }
}

// ---------------------------------------------------------------------------
// Kernel 2: flash attention per (b, 64-row L tile): O = softmax(QK^T/8)V,
// stores O head-major and the per-row (max, sumexp) stats for kernel 3.
// ---------------------------------------------------------------------------
__global__ __launch_bounds__(128)
void flash_kernel(const float* __restrict__ q_hm, const float* __restrict__ k_hm,
                  const float* __restrict__ v_hm, float* __restrict__ o_hm,
                  float* __restrict__ Mws, float* __restrict__ Lws) {
  __shared__ __align__(16) float sQ [64 * LDSS];
  __shared__ __align__(16) float sKP[64 * LDSS];   // K tile, then reused as P tile
  __shared__ __align__(16) float sV [64 * LDSS];

  const int b  = blockIdx.x;          // 0..31
  const int l0 = blockIdx.y * 64;
  const int tid  = threadIdx.x;
  const int lane = tid & 31;
  const int wv   = tid >> 5;
  const int l15  = lane & 15;
  const int hi   = lane >> 4;
  const int row  = tid >> 1;

  // stage Q tile (64x64)
  {
    const float* Qg = q_hm + ((size_t)b * LQ + l0) * HD;
#pragma unroll
    for (int i = 0; i < 8; ++i) {
      const int c = (tid & 1) * 32 + i * 4;
      *(float4*)&sQ[row * LDSS + c] = *(const float4*)&Qg[row * HD + c];
    }
  }
  __syncthreads();

  // preload this wave's Q A-fragments for all 16 k-steps (rows wv*16..+15)
  v2f qa[16];
#pragma unroll
  for (int ks = 0; ks < 16; ++ks)
    qa[ks] = *(const v2f*)&sQ[(wv * 16 + l15) * LDSS + ks * 4 + 2 * hi];

  v8f oacc[4];
#pragma unroll
  for (int j = 0; j < 4; ++j) oacc[j] = v8zero();
  float mrow[8], lrow[8];
#pragma unroll
  for (int r = 0; r < 8; ++r) { mrow[r] = -3.0e38f; lrow[r] = 0.f; }

  for (int s0 = 0; s0 < LQ; s0 += 64) {
    __syncthreads();  // previous chunk's P@V done before restaging
    const float* Kg = k_hm + ((size_t)b * LQ + s0) * HD;
    const float* Vg = v_hm + ((size_t)b * LQ + s0) * HD;
#pragma unroll
    for (int i = 0; i < 8; ++i) {
      const int c = (tid & 1) * 32 + i * 4;
      *(float4*)&sKP[row * LDSS + c] = *(const float4*)&Kg[row * HD + c];
      *(float4*)&sV [row * LDSS + c] = *(const float4*)&Vg[row * HD + c];
    }
    __syncthreads();

    // scores tile: this wave's 16 rows x 64 cols
    v8f sc[4];
#pragma unroll
    for (int j = 0; j < 4; ++j) sc[j] = v8zero();
#pragma unroll
    for (int ks = 0; ks < 16; ++ks) {
      v2f a = qa[ks];
#pragma unroll
      for (int j = 0; j < 4; ++j) {
        // B[k=d][n=s_local]: K row = s_local, contiguous float2 in d
        v2f bf = *(const v2f*)&sKP[(j * 16 + l15) * LDSS + ks * 4 + 2 * hi];
        sc[j] = wmma4(a, bf, sc[j]);
      }
    }
    // scale 1/sqrt(64)
#pragma unroll
    for (int j = 0; j < 4; ++j)
#pragma unroll
      for (int r = 0; r < 8; ++r) sc[j][r] *= 0.125f;

    // per-row chunk max (row = wave-local hi*8+r), reduce over 16 lanes
    float cmax[8];
#pragma unroll
    for (int r = 0; r < 8; ++r)
      cmax[r] = fmaxf(fmaxf(sc[0][r], sc[1][r]), fmaxf(sc[2][r], sc[3][r]));
#pragma unroll
    for (int off = 1; off < 16; off <<= 1)
#pragma unroll
      for (int r = 0; r < 8; ++r)
        cmax[r] = fmaxf(cmax[r], __shfl_xor(cmax[r], off, 32));

    float sfac[8];
#pragma unroll
    for (int r = 0; r < 8; ++r) {
      const float mnew = fmaxf(mrow[r], cmax[r]);
      sfac[r] = __expf(mrow[r] - mnew);
      mrow[r] = mnew;
    }
    // exponentiate
#pragma unroll
    for (int j = 0; j < 4; ++j)
#pragma unroll
      for (int r = 0; r < 8; ++r) sc[j][r] = __expf(sc[j][r] - mrow[r]);

    float csum[8];
#pragma unroll
    for (int r = 0; r < 8; ++r) csum[r] = sc[0][r] + sc[1][r] + sc[2][r] + sc[3][r];
#pragma unroll
    for (int off = 1; off < 16; off <<= 1)
#pragma unroll
      for (int r = 0; r < 8; ++r) csum[r] += __shfl_xor(csum[r], off, 32);

#pragma unroll
    for (int r = 0; r < 8; ++r) lrow[r] = lrow[r] * sfac[r] + csum[r];
#pragma unroll
    for (int j = 0; j < 4; ++j)
#pragma unroll
      for (int r = 0; r < 8; ++r) oacc[j][r] *= sfac[r];

    __syncthreads();  // everyone done reading K before overwriting with P
    // write P tile (wave-local 16 rows)
#pragma unroll
    for (int j = 0; j < 4; ++j)
#pragma unroll
      for (int r = 0; r < 8; ++r)
        sKP[(wv * 16 + hi * 8 + r) * LDSS + j * 16 + l15] = sc[j][r];

    // O += P @ V
#pragma unroll
    for (int ks = 0; ks < 16; ++ks) {
      v2f a = *(const v2f*)&sKP[(wv * 16 + l15) * LDSS + ks * 4 + 2 * hi];
#pragma unroll
      for (int j = 0; j < 4; ++j) {
        v2f bf;
        bf.x = sV[(ks * 4 + 2 * hi    ) * LDSS + j * 16 + l15];
        bf.y = sV[(ks * 4 + 2 * hi + 1) * LDSS + j * 16 + l15];
        oacc[j] = wmma4(a, bf, oacc[j]);
      }
    }
  }

  // finalize: O /= l ; write head-major O and the stats
  float* Og = o_hm + ((size_t)b * LQ + l0) * HD;
#pragma unroll
  for (int j = 0; j < 4; ++j)
#pragma unroll
    for (int r = 0; r < 8; ++r) {
      const int rr = wv * 16 + hi * 8 + r;
      Og[rr * HD + j * 16 + l15] = oacc[j][r] / lrow[r];
    }
  if (l15 == 0) {
#pragma unroll
    for (int r = 0; r < 8; ++r) {
      const int rr = wv * 16 + hi * 8 + r;
      Mws[b * LQ + l0 + rr] = mrow[r];
      Lws[b * LQ + l0 + rr] = lrow[r];
    }
  }
}

// ---------------------------------------------------------------------------
// Kernel 3: attention weights, mean over heads.
// block owns (n, 64 L rows, 64 S cols); loops h, recomputing scores via WMMA.
// ---------------------------------------------------------------------------
__global__ __launch_bounds__(128)
void weights_kernel(const float* __restrict__ q_hm, const float* __restrict__ k_hm,
                    const float* __restrict__ Mws, const float* __restrict__ Lws,
                    float* __restrict__ wout) {
  __shared__ __align__(16) float sQ[64 * LDSS];
  __shared__ __align__(16) float sK[64 * LDSS];

  const int s0 = blockIdx.x * 64;
  const int l0 = blockIdx.y * 64;
  const int n  = blockIdx.z;
  const int tid  = threadIdx.x;
  const int lane = tid & 31;
  const int wv   = tid >> 5;
  const int l15  = lane & 15;
  const int hi   = lane >> 4;
  const int row  = tid >> 1;

  float acc[4][8];
#pragma unroll
  for (int j = 0; j < 4; ++j)
#pragma unroll
    for (int r = 0; r < 8; ++r) acc[j][r] = 0.f;

  for (int h = 0; h < NH; ++h) {
    const int b = n * NH + h;
    __syncthreads();
    const float* Qg = q_hm + ((size_t)b * LQ + l0) * HD;
    const float* Kg = k_hm + ((size_t)b * LQ + s0) * HD;
#pragma unroll
    for (int i = 0; i < 8; ++i) {
      const int c = (tid & 1) * 32 + i * 4;
      *(float4*)&sQ[row * LDSS + c] = *(const float4*)&Qg[row * HD + c];
      *(float4*)&sK[row * LDSS + c] = *(const float4*)&Kg[row * HD + c];
    }
    __syncthreads();

    v8f sc[4];
#pragma unroll
    for (int j = 0; j < 4; ++j) sc[j] = v8zero();
#pragma unroll
    for (int ks = 0; ks < 16; ++ks) {
      v2f a = *(const v2f*)&sQ[(wv * 16 + l15) * LDSS + ks * 4 + 2 * hi];
#pragma unroll
      for (int j = 0; j < 4; ++j) {
        v2f bf = *(const v2f*)&sK[(j * 16 + l15) * LDSS + ks * 4 + 2 * hi];
        sc[j] = wmma4(a, bf, sc[j]);
      }
    }

    float Mv[8], rLv[8];
#pragma unroll
    for (int r = 0; r < 8; ++r) {
      const int rr = l0 + wv * 16 + hi * 8 + r;
      Mv[r]  = Mws[b * LQ + rr];
      rLv[r] = 1.0f / Lws[b * LQ + rr];
    }
#pragma unroll
    for (int j = 0; j < 4; ++j)
#pragma unroll
      for (int r = 0; r < 8; ++r)
        acc[j][r] += __expf(sc[j][r] * 0.125f - Mv[r]) * rLv[r];
  }

#pragma unroll
  for (int j = 0; j < 4; ++j)
#pragma unroll
    for (int r = 0; r < 8; ++r) {
      const int rr = l0 + wv * 16 + hi * 8 + r;
      wout[((size_t)n * LQ + rr) * LQ + s0 + j * 16 + l15] = acc[j][r] * (1.0f / NH);
    }
}

// ---------------------------------------------------------------------------
// Kernel 4: output projection: out[m=l*2+n][e] = (A @ Wo^T)[m][e] + bo[e],
// A gathered from head-major o_hm during LDS staging.
// ---------------------------------------------------------------------------
__global__ __launch_bounds__(128)
void oproj_kernel(const float* __restrict__ o_hm, const float* __restrict__ W,
                  const float* __restrict__ bias, float* __restrict__ out) {
  __shared__ __align__(16) float sA[64 * LDSS];
  __shared__ __align__(16) float sB[64 * LDSS];

  const int tid  = threadIdx.x;
  const int lane = tid & 31;
  const int wv   = tid >> 5;
  const int l15  = lane & 15;
  const int hi   = lane >> 4;
  const int m0   = blockIdx.x * 64;
  const int n0   = blockIdx.y * 64;
  const int wm   = (wv >> 1) * 32;
  const int wn   = (wv & 1) * 32;

  v8f acc[2][2];
  acc[0][0] = v8zero(); acc[0][1] = v8zero();
  acc[1][0] = v8zero(); acc[1][1] = v8zero();

  const int row   = tid >> 1;
  const int cbase = (tid & 1) * 16;
  const int m = m0 + row, l = m >> 1, nn = m & 1;

  for (int k0 = 0; k0 < EM; k0 += 32) {
    __syncthreads();
#pragma unroll
    for (int i = 0; i < 4; ++i) {
      const int c = cbase + i * 4;
      const int k = k0 + c, h = k >> 6, d = k & 63;
      *(float4*)&sA[row * LDSS + c] =
          *(const float4*)&o_hm[((size_t)(nn * NH + h) * LQ + l) * HD + d];
      *(float4*)&sB[row * LDSS + c] = *(const float4*)&W[(n0 + row) * EM + k];
    }
    if (k0 + 32 < EM)
      __builtin_prefetch(&W[(n0 + row) * EM + k0 + 32 + cbase], 0, 3);
    __syncthreads();
#pragma unroll
    for (int kk = 0; kk < 32; kk += 4) {
      v2f a0 = *(const v2f*)&sA[(wm      + l15) * LDSS + kk + 2 * hi];
      v2f a1 = *(const v2f*)&sA[(wm + 16 + l15) * LDSS + kk + 2 * hi];
      v2f b0 = *(const v2f*)&sB[(wn      + l15) * LDSS + kk + 2 * hi];
      v2f b1 = *(const v2f*)&sB[(wn + 16 + l15) * LDSS + kk + 2 * hi];
      acc[0][0] = wmma4(a0, b0, acc[0][0]);
      acc[0][1] = wmma4(a0, b1, acc[0][1]);
      acc[1][0] = wmma4(a1, b0, acc[1][0]);
      acc[1][1] = wmma4(a1, b1, acc[1][1]);
    }
  }

#pragma unroll
  for (int im = 0; im < 2; ++im) {
#pragma unroll
    for (int in = 0; in < 2; ++in) {
      const int e = n0 + wn + in * 16 + l15;
      const float bv = bias[e];
#pragma unroll
      for (int r = 0; r < 8; ++r) {
        const int mm = m0 + wm + im * 16 + hi * 8 + r;
        out[(size_t)mm * EM + e] = acc[im][in][r] + bv;
      }
    }
  }
}

// ---------------------------------------------------------------------------
extern "C" void kernel_launch(void* const* d_in, const int* in_sizes, int n_in,
                              void* d_out, int out_size, void* d_ws, size_t ws_size,
                              hipStream_t stream) {
  const float* query = (const float*)d_in[0];
  const float* key   = (const float*)d_in[1];
  const float* value = (const float*)d_in[2];
  const float* Wq = (const float*)d_in[3]; const float* bq = (const float*)d_in[4];
  const float* Wk = (const float*)d_in[5]; const float* bk = (const float*)d_in[6];
  const float* Wv = (const float*)d_in[7]; const float* bv = (const float*)d_in[8];
  const float* Wo = (const float*)d_in[9]; const float* bo = (const float*)d_in[10];

  float* out = (float*)d_out;
  float* ws  = (float*)d_ws;

  const size_t HM = (size_t)NBH * LQ * HD;   // 4,194,304 floats per tensor
  float* q_hm = ws;
  float* k_hm = ws + HM;
  float* v_hm = ws + 2 * HM;
  float* o_hm = ws + 3 * HM;
  float* Mws  = ws + 4 * HM;                 // 65,536 floats
  float* Lws  = ws + 4 * HM + (size_t)NBH * LQ;

  float* attn_out = out;                     // [2048, 2, 1024]
  float* attn_w   = out + (size_t)LQ * NB * EM;  // [2, 2048, 2048]

  dim3 blk(128);
  dim3 gproj(64, 16);              // (4096/64) x (1024/64)
  proj_kernel<<<gproj, blk, 0, stream>>>(query, Wq, bq, q_hm);
  proj_kernel<<<gproj, blk, 0, stream>>>(key,   Wk, bk, k_hm);
  proj_kernel<<<gproj, blk, 0, stream>>>(value, Wv, bv, v_hm);

  flash_kernel<<<dim3(NBH, LQ / 64), blk, 0, stream>>>(q_hm, k_hm, v_hm, o_hm, Mws, Lws);

  weights_kernel<<<dim3(LQ / 64, LQ / 64, NB), blk, 0, stream>>>(q_hm, k_hm, Mws, Lws, attn_w);

  oproj_kernel<<<gproj, blk, 0, stream>>>(o_hm, Wo, bo, attn_out);
}